// TransformedInput_19104014532646
// MI455X (gfx1250) — compile-verified
//
#include <hip/hip_runtime.h>
#include <stdint.h>

// Problem constants from the reference: SHAPE = (1,1,96,96), EPS = 0.1
#define ZN      9216          // N = C*H*W
#define ZEPS    0.1f
#define ZTHREADS 256
#define ZPER    36            // ZN / ZTHREADS
#define ZOUT_ELEMS ((1 + ZN) * ZN)     // 84,943,872 floats (~340 MB)
#define ZOUT_V4    (ZOUT_ELEMS / 4)    // 21,235,968 16-byte chunks

typedef __attribute__((ext_vector_type(4))) float v4f;

// ---------------------------------------------------------------------------
// Kernel 1: per-pixel zonotope math + block-wide stream-compaction scan.
// One block of 256 threads (8 wave32 waves), 36 pixels per thread.
// (Round-1 codegen: s_clause'd global_load_b128, v_pk_fma_f32 packed math,
//  ds_* LDS scan, global_prefetch_b8 — keep as-is.)
// ---------------------------------------------------------------------------
__global__ void zono_compute_scan(const float* __restrict__ x,
                                  float* __restrict__ bias,
                                  float* __restrict__ ef,
                                  int* __restrict__ slot) {
  __shared__ int counts[ZTHREADS];
  const int t = threadIdx.x;
  const int base = t * ZPER;

  __builtin_prefetch(&x[base], 0, 3);   // global_prefetch_b8

  int cnt = 0;
  float e_loc[ZPER];
  for (int k = 0; k < ZPER; ++k) {
    const float xv = x[base + k];
    const float rl = fmaxf(ZEPS - xv, 0.0f) * 0.5f;
    const float rh = fmaxf(xv - (1.0f - ZEPS), 0.0f) * 0.5f;
    const float b  = xv + rl - rh;            // zonotope center
    const float e  = ZEPS - rl - rh;          // error magnitude
    bias[base + k] = b;
    ef[base + k]   = e;
    e_loc[k] = e;
    cnt += (e >= 0.0f) ? 1 : 0;
  }
  counts[t] = cnt;
  __syncthreads();

  // Exclusive scan of 256 per-thread counts (tiny; serial LDS reads are fine)
  int pre = 0;
  for (int i = 0; i < t; ++i) pre += counts[i];

  int s = pre;
  for (int k = 0; k < ZPER; ++k) {
    if (e_loc[k] >= 0.0f) {
      s += 1;
      slot[base + k] = s;
    } else {
      slot[base + k] = -1;
    }
  }
}

// ---------------------------------------------------------------------------
// Kernel 2: stream ~340 MB of zeros via the CDNA5 async data mover:
// GLOBAL_STORE_ASYNC_FROM_LDS_B128 (ASYNCcnt-tracked LDS->global DMA path).
// Each instruction moves 16 B/lane = 512 B/wave straight from an LDS zero
// tile — same per-issue bandwidth as a VGPR-sourced global_store_b128, but
// on the dedicated async path with no VGPR data staging.  HW stalls issue if
// ASYNCcnt (6-bit) would overflow, so the unbounded issue loop is safe; one
// s_wait_asynccnt 0 drains before exit.  Roofline: 340 MB / 23.3 TB/s ~ 15 us.
// ---------------------------------------------------------------------------
__global__ void zono_fill_async(float* __restrict__ out) {
  // 2 KB zero tile: 128 distinct 16-byte chunks so lanes read different banks.
  __shared__ __align__(16) float ztile[512];
  for (int i = threadIdx.x; i < 512; i += ZTHREADS) ztile[i] = 0.0f;
  __syncthreads();

  // Wave-relative LDS byte offset of this lane's 16B zero chunk
  // (explicit addrspace(3) cast: as(3) pointers are 32-bit LDS offsets).
  const uint32_t lds_off = (uint32_t)(size_t)
      (__attribute__((address_space(3))) void*)&ztile[(threadIdx.x & 127) * 4];

  const uint64_t base = (uint64_t)out;
  const size_t stride = (size_t)gridDim.x * blockDim.x;
  size_t i = (size_t)blockIdx.x * blockDim.x + threadIdx.x;

  for (; i < (size_t)ZOUT_V4; i += stride) {
    const uint64_t ga = base + (i << 4);   // byte address of 16B chunk
    asm volatile("global_store_async_from_lds_b128 %0, %1, off"
                 :: "v"(ga), "v"(lds_off)
                 : "memory");
  }
  asm volatile("s_wait_asynccnt 0x0" ::: "memory");
}

// ---------------------------------------------------------------------------
// Kernel 3: slot-0 bias image + one scattered error term per qualifying pixel.
// 9216 threads; runs after the fill (stream ordering).
// ---------------------------------------------------------------------------
__global__ void zono_scatter(const float* __restrict__ bias,
                             const float* __restrict__ ef,
                             const int* __restrict__ slot,
                             float* __restrict__ out) {
  const int n = (int)(blockIdx.x * blockDim.x + threadIdx.x);
  if (n >= ZN) return;
  out[n] = bias[n];                       // zono[0, 0, :, :, :]
  const int s = slot[n];
  if (s >= 0) {
    out[(size_t)s * ZN + n] = ef[n];      // zono[0, s, f, i, j] diagonal term
  }
}

extern "C" void kernel_launch(void* const* d_in, const int* in_sizes, int n_in,
                              void* d_out, int out_size, void* d_ws, size_t ws_size,
                              hipStream_t stream) {
  (void)in_sizes; (void)n_in; (void)out_size; (void)ws_size;

  const float* x = (const float*)d_in[0];
  float* out = (float*)d_out;

  // Workspace layout: bias[N] | ef[N] | slot[N]  (108 KB total)
  float* ws_bias = (float*)d_ws;
  float* ws_ef   = ws_bias + ZN;
  int*   ws_slot = (int*)(ws_ef + ZN);

  // 1) elementwise + scan (single block, 8 waves)
  zono_compute_scan<<<1, ZTHREADS, 0, stream>>>(x, ws_bias, ws_ef, ws_slot);

  // 2) zero-fill 340 MB via async LDS->global b128 stores (dominant cost)
  zono_fill_async<<<8192, ZTHREADS, 0, stream>>>(out);

  // 3) bias row + sparse error scatter (overwrites a few KB after the fill)
  zono_scatter<<<(ZN + ZTHREADS - 1) / ZTHREADS, ZTHREADS, 0, stream>>>(
      ws_bias, ws_ef, ws_slot, out);
}